// StabilizedAudioLayer_48808008351961
// MI455X (gfx1250) — compile-verified
//
#include <hip/hip_runtime.h>

// ---------------------------------------------------------------------------
// 512-point FFT per window on WMMA f32 16x16x4 (CDNA5 / gfx1250, wave32).
// Four-step: n = n1 + 32*n2, k = k2 + 16*k1f
//   step1: Y[n1][k2]  = sum_n2 x[n1+32*n2] * W16^{n2*k2}      (32x16 @ 16x16 matmul)
//   step2: Y *= W512^{n1*k2}
//   step3: DFT-32 over n1 per k2 via even/odd 16x16 matmuls + radix-2 twiddle
// Inverse is the same structure reversed with conjugated constants, 1/512.
// All trig (DFT-16 fragment constants + twiddles) is lane-invariant -> hoisted
// into per-thread registers once per kernel and reused across 4 windows/wave.
// ---------------------------------------------------------------------------

typedef __attribute__((ext_vector_type(2))) float v2f;
typedef __attribute__((ext_vector_type(8))) float v8f;

#define WIN 512
#define WAVES_PER_BLOCK 4
#define TPB (WAVES_PER_BLOCK * 32)
#define ITER 4  // windows per wave
#define MTPB 256

static __device__ __forceinline__ int laneid() { return threadIdx.x & 31; }

static __device__ __forceinline__ v8f wmma4(v2f a, v2f b, v8f c) {
  // (neg_a, A, neg_b, B, c_mod, C, reuse_a, reuse_b); f32 WMMA: A/B neg must be 0
  return __builtin_amdgcn_wmma_f32_16x16x4_f32(false, a, false, b, (short)0, c,
                                               false, false);
}

// ---- data fragment builders ---------------------------------------------
// A operand (16x4): lanes 0-15 -> M=lane, K=k0{+0,+1}; lanes 16-31 -> K=k0{+2,+3}
static __device__ __forceinline__ v2f fragA_lds(const float* m, int rowOff, int k0) {
  int l = laneid();
  int r = rowOff + (l & 15);
  int k = k0 + ((l >> 4) << 1);
  v2f a;
  a.x = m[r * 16 + k];
  a.y = m[r * 16 + k + 1];
  return a;
}
// B operand (4x16): row K (mapped row = rowOff + 2*K), col N = lane&15
static __device__ __forceinline__ v2f fragB_lds(const float* m, int rowOff, int k0) {
  int l = laneid();
  int n = l & 15;
  int k = k0 + ((l >> 4) << 1);
  v2f b;
  b.x = m[(rowOff + 2 * k) * 16 + n];
  b.y = m[(rowOff + 2 * (k + 1)) * 16 + n];
  return b;
}

// ---- hoisted per-thread constants ---------------------------------------
// DFT-16 fragment constants: A-op value trig(2pi*(lane&15)*K/16) equals the
// B-op value (same index product), so one set serves both operand roles.
struct Consts {
  v2f dC[4], dS[4], dNS[4];     // cos / sin / -sin, K-chunks 0,4,8,12
  float cc[2][8], ss[2][8];     // cos/sin(2pi*n1*k2/512), n1 = blk*16+8*(l>>4)+r
  float c32[8], s32[8];         // cos/sin(2pi*k1/32),     k1 = 8*(l>>4)+r
};

static __device__ void init_consts(Consts& K) {
  int l = laneid();
  int mn = l & 15;
#pragma unroll
  for (int j = 0; j < 4; ++j) {
    int k = 4 * j + ((l >> 4) << 1);
    float s0, c0, s1, c1;
    __sincosf(0.39269908169872414f * (float)((mn * k) & 15), &s0, &c0);
    __sincosf(0.39269908169872414f * (float)((mn * (k + 1)) & 15), &s1, &c1);
    K.dC[j].x = c0;  K.dC[j].y = c1;
    K.dS[j].x = s0;  K.dS[j].y = s1;
    K.dNS[j].x = -s0; K.dNS[j].y = -s1;
  }
  int k2 = mn;
#pragma unroll
  for (int blk = 0; blk < 2; ++blk)
#pragma unroll
    for (int r = 0; r < 8; ++r) {
      int n1 = blk * 16 + ((l >> 4) << 3) + r;
      float s, c;
      __sincosf(0.012271846303085129f * (float)(n1 * k2), &s, &c);
      K.cc[blk][r] = c;
      K.ss[blk][r] = s;
    }
#pragma unroll
  for (int r = 0; r < 8; ++r) {
    int k1 = ((l >> 4) << 3) + r;
    float s, c;
    __sincosf(0.19634954084936207f * (float)k1, &s, &c);
    K.c32[r] = c;
    K.s32[r] = s;
  }
}

// ---- 16x16x16 matmul chains (4 chained K=4 WMMAs) -----------------------
static __device__ __forceinline__ v8f mm_Alds_Bc(const float* A, int rowOff,
                                                 const v2f* B4, v8f acc) {
#pragma unroll
  for (int j = 0; j < 4; ++j)
    acc = wmma4(fragA_lds(A, rowOff, 4 * j), B4[j], acc);
  return acc;
}
static __device__ __forceinline__ v8f mm_Ac_Blds(const v2f* A4, const float* B,
                                                 int rowOff, v8f acc) {
#pragma unroll
  for (int j = 0; j < 4; ++j)
    acc = wmma4(A4[j], fragB_lds(B, rowOff, 4 * j), acc);
  return acc;
}

struct Spec {      // per-wave spectrum fragments, C/D layout:
  v8f pr, pi;      // k1 in [0,16)  : element r -> M = r + 8*(lane>>4) = k1, N = k2
  v8f mr, mi;      // k1 in [16,32)
};

// Forward FFT. px holds x in (n1,n2) layout: px[n1*16+n2] = x[n1+32*n2].
// yr/yi are per-wave scratch planes. Block-uniform; contains one barrier.
static __device__ Spec fft512_fwd(const Consts& K, const float* px, float* yr,
                                  float* yi) {
  int l = laneid();
  int k2 = l & 15;
#pragma unroll
  for (int blk = 0; blk < 2; ++blk) {
    v8f ar = {}, ai = {};
    ar = mm_Alds_Bc(px, blk * 16, K.dC, ar);    // A * cos
    ai = mm_Alds_Bc(px, blk * 16, K.dNS, ai);   // A * (-sin)
#pragma unroll
    for (int r = 0; r < 8; ++r) {               // * exp(-i*2pi*n1*k2/512)
      int n1 = blk * 16 + ((l >> 4) << 3) + r;
      float c = K.cc[blk][r], s = K.ss[blk][r];
      yr[n1 * 16 + k2] = ar[r] * c + ai[r] * s;
      yi[n1 * 16 + k2] = ai[r] * c - ar[r] * s;
    }
  }
  __syncthreads();
  // step3: E = F16(-) x Y_even, O = F16(-) x Y_odd  (F = C - iS)
  v8f er = {}, ei = {}, og = {}, oi = {};
  er = mm_Ac_Blds(K.dC, yr, 0, er);    //  C*Yer
  er = mm_Ac_Blds(K.dS, yi, 0, er);    // +S*Yei
  ei = mm_Ac_Blds(K.dC, yi, 0, ei);    //  C*Yei
  ei = mm_Ac_Blds(K.dNS, yr, 0, ei);   // -S*Yer
  og = mm_Ac_Blds(K.dC, yr, 1, og);
  og = mm_Ac_Blds(K.dS, yi, 1, og);
  oi = mm_Ac_Blds(K.dC, yi, 1, oi);
  oi = mm_Ac_Blds(K.dNS, yr, 1, oi);
  Spec sp;
#pragma unroll
  for (int r = 0; r < 8; ++r) {        // radix-2: w = exp(-i*2pi*k1/32)
    float c = K.c32[r], s = K.s32[r];
    float tr = og[r] * c + oi[r] * s;
    float ti = oi[r] * c - og[r] * s;
    sp.pr[r] = er[r] + tr;
    sp.pi[r] = ei[r] + ti;
    sp.mr[r] = er[r] - tr;
    sp.mi[r] = ei[r] - ti;
  }
  return sp;
}

// Inverse FFT of fragment spectrum; writes real part * 1/512 to out (if non-null).
// zr/zi are per-wave scratch. Caller must __syncthreads() before calling.
static __device__ void fft512_inv_store(const Consts& K, Spec sp, float* zr,
                                        float* zi, float* out) {
  int l = laneid();
  int k2 = l & 15;
#pragma unroll
  for (int r = 0; r < 8; ++r) {      // stage Z (rows = k1full)
    int k1 = ((l >> 4) << 3) + r;
    zr[k1 * 16 + k2] = sp.pr[r];
    zi[k1 * 16 + k2] = sp.pi[r];
    zr[(k1 + 16) * 16 + k2] = sp.mr[r];
    zi[(k1 + 16) * 16 + k2] = sp.mi[r];
  }
  __syncthreads();
  // stepA: E' = F16(+) x Z_even, O' = F16(+) x Z_odd  (G = C + iS)
  v8f er = {}, ei = {}, og = {}, oi = {};
  er = mm_Ac_Blds(K.dC, zr, 0, er);    //  C*Zer
  er = mm_Ac_Blds(K.dNS, zi, 0, er);   // -S*Zei
  ei = mm_Ac_Blds(K.dC, zi, 0, ei);    //  C*Zei
  ei = mm_Ac_Blds(K.dS, zr, 0, ei);    // +S*Zer
  og = mm_Ac_Blds(K.dC, zr, 1, og);
  og = mm_Ac_Blds(K.dNS, zi, 1, og);
  oi = mm_Ac_Blds(K.dC, zi, 1, oi);
  oi = mm_Ac_Blds(K.dS, zr, 1, oi);
  // combine (wc = exp(+i*2pi*n1/32)) and stepB twiddle exp(+i*2pi*n1*k2/512)
  float upr[8], upi[8], umr[8], umi[8];
#pragma unroll
  for (int r = 0; r < 8; ++r) {
    float c = K.c32[r], s = K.s32[r];
    float tr = og[r] * c - oi[r] * s;
    float ti = oi[r] * c + og[r] * s;
    float ar = er[r] + tr, ai = ei[r] + ti;   // U[n1]
    float br = er[r] - tr, bi = ei[r] - ti;   // U[n1+16]
    float c1 = K.cc[0][r], s1 = K.ss[0][r];
    float c2 = K.cc[1][r], s2 = K.ss[1][r];
    upr[r] = ar * c1 - ai * s1;
    upi[r] = ai * c1 + ar * s1;
    umr[r] = br * c2 - bi * s2;
    umi[r] = bi * c2 + br * s2;
  }
  __syncthreads();                 // stepA reads done before overwriting planes
#pragma unroll
  for (int r = 0; r < 8; ++r) {
    int n1 = ((l >> 4) << 3) + r;
    zr[n1 * 16 + k2] = upr[r];
    zi[n1 * 16 + k2] = upi[r];
    zr[(n1 + 16) * 16 + k2] = umr[r];
    zi[(n1 + 16) * 16 + k2] = umi[r];
  }
  __syncthreads();
  // stepC: V = U x G2 (G2 = exp(+i*2pi*k2*n2/16)); real part only
#pragma unroll
  for (int blk = 0; blk < 2; ++blk) {
    v8f vr = {};
    vr = mm_Alds_Bc(zr, blk * 16, K.dC, vr);    //  Ur*C
    vr = mm_Alds_Bc(zi, blk * 16, K.dNS, vr);   // -Ui*S
    if (out) {
      // element r: n = (blk*16 + 8*(lane>>4) + 32*(lane&15)) + r -> contiguous
      int base = blk * 16 + ((l >> 4) << 3) + 32 * (l & 15);
      float4 lo, hi;
      lo.x = vr[0] * (1.0f / 512.0f);
      lo.y = vr[1] * (1.0f / 512.0f);
      lo.z = vr[2] * (1.0f / 512.0f);
      lo.w = vr[3] * (1.0f / 512.0f);
      hi.x = vr[4] * (1.0f / 512.0f);
      hi.y = vr[5] * (1.0f / 512.0f);
      hi.z = vr[6] * (1.0f / 512.0f);
      hi.w = vr[7] * (1.0f / 512.0f);
      reinterpret_cast<float4*>(out + base)[0] = lo;
      reinterpret_cast<float4*>(out + base)[1] = hi;
    }
  }
}

// coalesced float4 load of one 512-sample window -> (n1,n2) LDS layout
static __device__ __forceinline__ void stage_window(const float* src, float* px,
                                                    bool valid) {
  int l = laneid();
#pragma unroll
  for (int j = 0; j < 4; ++j) {
    float4 v = valid ? reinterpret_cast<const float4*>(src)[l * 4 + j]
                     : make_float4(0.f, 0.f, 0.f, 0.f);
    int p = l * 16 + j * 4;
    px[((p + 0) & 31) * 16 + ((p + 0) >> 5)] = v.x;
    px[((p + 1) & 31) * 16 + ((p + 1) >> 5)] = v.y;
    px[((p + 2) & 31) * 16 + ((p + 2) >> 5)] = v.z;
    px[((p + 3) & 31) * 16 + ((p + 3) >> 5)] = v.w;
  }
}

// ---------------------------------------------------------------------------
// Pass A: forward FFT per window, accumulate freq energy (LDS -> global atomic),
// write per-window time energy (stashed in d_out; pass B overwrites d_out).
// ---------------------------------------------------------------------------
__global__ void __launch_bounds__(TPB)
fft_energy_kernel(const float* __restrict__ x, float* __restrict__ freqE,
                  float* __restrict__ timeE, int T) {
  __shared__ float planes[WAVES_PER_BLOCK][3][WIN];
  __shared__ float fE[WIN];
  Consts K;
  init_consts(K);
  int wave = threadIdx.x >> 5;
  int l = threadIdx.x & 31;
  for (int i = threadIdx.x; i < WIN; i += TPB) fE[i] = 0.0f;
  float* px = planes[wave][0];
  float* p1 = planes[wave][1];
  float* p2 = planes[wave][2];
  int base_row = (blockIdx.x * WAVES_PER_BLOCK + wave) * ITER;
#pragma unroll 1
  for (int it = 0; it < ITER; ++it) {
    int row = base_row + it;
    bool valid = row < T;
    stage_window(x + (size_t)row * WIN, px, valid);
    __syncthreads();
    Spec sp = fft512_fwd(K, px, p1, p2);
    float tsum = 0.0f;
#pragma unroll
    for (int r = 0; r < 8; ++r) {
      int k1 = ((l >> 4) << 3) + r;
      int k2 = l & 15;
      float ep = sp.pr[r] * sp.pr[r] + sp.pi[r] * sp.pi[r];
      float em = sp.mr[r] * sp.mr[r] + sp.mi[r] * sp.mi[r];
      tsum += ep + em;
      if (valid) {
        atomicAdd(&fE[k2 + 16 * k1], ep);       // ds_add_f32
        atomicAdd(&fE[k2 + 16 * k1 + 256], em);
      }
    }
#pragma unroll
    for (int off = 16; off > 0; off >>= 1) tsum += __shfl_xor(tsum, off, 32);
    if (l == 0 && valid) timeE[row] = tsum;     // mean scale cancels in mask
    __syncthreads();
  }
  for (int i = threadIdx.x; i < WIN; i += TPB)
    unsafeAtomicAdd(&freqE[i], fE[i]);          // global_atomic_add_f32
}

// ---------------------------------------------------------------------------
// Mask selection: threshold-vs-max with exact k-th-largest fallback found by
// 31-step bitwise binary search on float bit patterns (monotone for e >= 0).
// Single block.
// ---------------------------------------------------------------------------
static __device__ float block_max(const float* d, int n, float* red) {
  int t = threadIdx.x;
  float m = 0.0f;
  for (int i = t; i < n; i += blockDim.x) m = fmaxf(m, d[i]);
  red[t] = m;
  __syncthreads();
  for (int s = blockDim.x >> 1; s > 0; s >>= 1) {
    if (t < s) red[t] = fmaxf(red[t], red[t + s]);
    __syncthreads();
  }
  float r = red[0];
  __syncthreads();
  return r;
}
static __device__ int block_count(const float* d, int n, float cut, bool ge,
                                  int* redi) {
  int t = threadIdx.x;
  int c = 0;
  for (int i = t; i < n; i += blockDim.x)
    c += (ge ? (d[i] >= cut) : (d[i] > cut)) ? 1 : 0;
  redi[t] = c;
  __syncthreads();
  for (int s = blockDim.x >> 1; s > 0; s >>= 1) {
    if (t < s) redi[t] += redi[t + s];
    __syncthreads();
  }
  int r = redi[0];
  __syncthreads();
  return r;
}
static __device__ void make_mask(const float* e, int n, int min_keep, float thr,
                                 float* mask, int* redi) {
  int cnt = block_count(e, n, thr, false, redi);
  bool topk = cnt < min_keep;
  float cut = thr;
  if (topk) {
    unsigned p = 0u;
    for (int bit = 30; bit >= 0; --bit) {
      unsigned cand = p | (1u << bit);
      if (block_count(e, n, __uint_as_float(cand), true, redi) >= min_keep)
        p = cand;
    }
    cut = __uint_as_float(p);
  }
  for (int i = threadIdx.x; i < n; i += blockDim.x)
    mask[i] = (topk ? (e[i] >= cut) : (e[i] > thr)) ? 1.0f : 0.0f;
  __syncthreads();
}

__global__ void __launch_bounds__(MTPB)
select_mask_kernel(const float* __restrict__ freqE,
                   const float* __restrict__ timeE, float* __restrict__ fmask,
                   float* __restrict__ tmask, int T, int minF, int minT) {
  __shared__ float red[MTPB];
  __shared__ int redi[MTPB];
  float fm = block_max(freqE, WIN, red);
  make_mask(freqE, WIN, minF, 0.02f * fm, fmask, redi);
  float tm = block_max(timeE, T, red);
  make_mask(timeE, T, minT, 0.02f * tm, tmask, redi);
}

// ---------------------------------------------------------------------------
// Pass B: forward FFT (recompute), apply time x freq mask, inverse FFT, store.
// smax normalization cancels end-to-end (all ops linear, masks scale-invariant).
// ---------------------------------------------------------------------------
__global__ void __launch_bounds__(TPB)
fft_mask_ifft_kernel(const float* __restrict__ x,
                     const float* __restrict__ fmask,
                     const float* __restrict__ tmask, float* __restrict__ out,
                     int T) {
  __shared__ float planes[WAVES_PER_BLOCK][3][WIN];
  __shared__ float fm[WIN];
  Consts K;
  init_consts(K);
  int wave = threadIdx.x >> 5;
  int l = threadIdx.x & 31;
  for (int i = threadIdx.x; i < WIN; i += TPB) fm[i] = fmask[i];
  float* px = planes[wave][0];
  float* p1 = planes[wave][1];
  float* p2 = planes[wave][2];
  int base_row = (blockIdx.x * WAVES_PER_BLOCK + wave) * ITER;
#pragma unroll 1
  for (int it = 0; it < ITER; ++it) {
    int row = base_row + it;
    bool valid = row < T;
    float tmv = valid ? tmask[row] : 0.0f;
    stage_window(x + (size_t)row * WIN, px, valid);
    __syncthreads();
    Spec sp = fft512_fwd(K, px, p1, p2);
#pragma unroll
    for (int r = 0; r < 8; ++r) {
      int k1 = ((l >> 4) << 3) + r;
      int k2 = l & 15;
      float mp = tmv * fm[k2 + 16 * k1];
      float mm = tmv * fm[k2 + 16 * k1 + 256];
      sp.pr[r] *= mp;
      sp.pi[r] *= mp;
      sp.mr[r] *= mm;
      sp.mi[r] *= mm;
    }
    __syncthreads();  // step3 plane reads done before inverse overwrites them
    fft512_inv_store(K, sp, p1, p2,
                     valid ? out + (size_t)row * WIN : nullptr);
    __syncthreads();  // stepC plane reads done before next iter's step1 writes
  }
}

__global__ void zero_kernel(float* p, int n) {
  int i = blockIdx.x * blockDim.x + threadIdx.x;
  if (i < n) p[i] = 0.0f;
}

// ---------------------------------------------------------------------------
extern "C" void kernel_launch(void* const* d_in, const int* in_sizes, int n_in,
                              void* d_out, int out_size, void* d_ws,
                              size_t ws_size, hipStream_t stream) {
  (void)n_in;
  (void)out_size;
  (void)ws_size;
  const float* x = (const float*)d_in[0];
  float* out = (float*)d_out;
  int n = in_sizes[0];
  int T = n / WIN;  // 65536 for the 4096x8192 input (exact)

  float* ws = (float*)d_ws;
  float* freqE = ws;            // [512]
  float* fmask = ws + WIN;      // [512]
  float* tmask = ws + 2 * WIN;  // [T]
  float* timeE = out;           // stash time energies in d_out; pass B overwrites

  int minF = (int)(WIN * (1.0 - 0.3 * 0.8));
  if (minF < 1) minF = 1;
  int minT = (int)((double)T * (1.0 - 0.3 * 0.8));
  if (minT < 1) minT = 1;

  int rows_per_block = WAVES_PER_BLOCK * ITER;
  int nblk = (T + rows_per_block - 1) / rows_per_block;
  zero_kernel<<<(WIN + 255) / 256, 256, 0, stream>>>(freqE, WIN);
  fft_energy_kernel<<<nblk, TPB, 0, stream>>>(x, freqE, timeE, T);
  select_mask_kernel<<<1, MTPB, 0, stream>>>(freqE, timeE, fmask, tmask, T,
                                             minF, minT);
  fft_mask_ifft_kernel<<<nblk, TPB, 0, stream>>>(x, fmask, tmask, out, T);
}